// iBlock_90409061581006
// MI455X (gfx1250) — compile-verified
//
#include <hip/hip_runtime.h>
#include <hip/hip_bf16.h>
#include <math.h>

// Shapes (compile-time constants from the reference):
// B=8, PN=8 -> BP=64; PL=511 -> L=512; NV=2; H=8; DK=DV=32; D=256; DM=512
#define BP   64
#define L    512
#define PL   511
#define NHEAD 8
#define DK   32
#define DMOD 256          // N_HEAD*DV
#define DFF  1024         // 4*D
#define DM   512          // d_model

typedef __attribute__((ext_vector_type(2))) float v2f;
typedef __attribute__((ext_vector_type(8))) float v8f;

// ---------------------------------------------------------------------------
// Kernel 1: attention, reduced to the q=0 row.  One block per bp, wave==head.
// ---------------------------------------------------------------------------
__global__ __launch_bounds__(256) void attn_kernel(
    const float* __restrict__ x,      // [BP, PL, 2]
    const float* __restrict__ t,      // [BP, L]
    const int*   __restrict__ mask,   // [BP, L]
    const float* __restrict__ emb,    // [2]
    const float* __restrict__ Wq,     // [2, 256]
    const float* __restrict__ Wk,     // [2, 256]
    const float* __restrict__ Wv,     // [2, 256]
    float* __restrict__ last)         // [BP, 256]
{
    __shared__ float xs[L * 2];
    __shared__ float ts[L];
    __shared__ int   ms[L];
    __shared__ float wq[2 * DMOD];
    __shared__ float wk[2 * DMOD];
    __shared__ float wv[2 * DMOD];
    __shared__ float sc[NHEAD][L];

    const int bp  = blockIdx.x;
    const int tid = threadIdx.x;

    // Stage inputs into LDS.
    for (int i = tid; i < L; i += 256) {
        float a0, a1;
        if (i == 0) { a0 = emb[0]; a1 = emb[1]; }
        else {
            a0 = x[(bp * PL + (i - 1)) * 2 + 0];
            a1 = x[(bp * PL + (i - 1)) * 2 + 1];
        }
        xs[i * 2 + 0] = a0;
        xs[i * 2 + 1] = a1;
        ts[i] = t[bp * L + i];
        ms[i] = mask[bp * L + i];
    }
    for (int i = tid; i < 2 * DMOD; i += 256) {
        wq[i] = Wq[i]; wk[i] = Wk[i]; wv[i] = Wv[i];
    }
    __syncthreads();

    const int wave = tid >> 5;
    const int lane = tid & 31;
    const int h    = wave;               // 8 waves == 8 heads

    const float L2_1000 = 9.965784284662087f;   // log2(1000)

    // RoPE'd q0 for this head (all lanes redundantly hold it), pre-scaled.
    float q0[DK];
    {
        const float t0 = ts[0];
        const float e0 = xs[0], e1 = xs[1];
        for (int j = 0; j < DK / 2; ++j) {
            float theta = exp2f(-L2_1000 * (float)j * (1.0f / 16.0f));
            float qre = e0 * wq[h * 32 + 2 * j]     + e1 * wq[DMOD + h * 32 + 2 * j];
            float qim = e0 * wq[h * 32 + 2 * j + 1] + e1 * wq[DMOD + h * 32 + 2 * j + 1];
            float sn, cs;
            __sincosf(t0 * theta, &sn, &cs);
            q0[2 * j]     = qre * cs - qim * sn;
            q0[2 * j + 1] = qre * sn + qim * cs;
        }
        const float inv_scale = 0.17677669529663687f;  // 1/sqrt(DK)
        for (int d = 0; d < DK; ++d) q0[d] *= inv_scale;
    }

    const int m0 = ms[0];

    // Phase 1: scores over 512 keys (each lane owns 16 positions).
    for (int it = 0; it < L / 32; ++it) {
        int l = it * 32 + lane;
        float a0 = xs[l * 2], a1 = xs[l * 2 + 1];
        float tl = ts[l];
        float dot = 0.f;
        for (int j = 0; j < DK / 2; ++j) {
            float theta = exp2f(-L2_1000 * (float)j * (1.0f / 16.0f));
            float kre = a0 * wk[h * 32 + 2 * j]     + a1 * wk[DMOD + h * 32 + 2 * j];
            float kim = a0 * wk[h * 32 + 2 * j + 1] + a1 * wk[DMOD + h * 32 + 2 * j + 1];
            float sn, cs;
            __sincosf(tl * theta, &sn, &cs);
            dot += q0[2 * j] * (kre * cs - kim * sn) + q0[2 * j + 1] * (kre * sn + kim * cs);
        }
        if (m0 * ms[l] == 0) dot = -1e9f;   // finite "-INF" exactly like reference
        sc[h][l] = dot;
    }
    __syncthreads();

    // Phase 2: softmax per head (wave32 reductions).
    float mx = -INFINITY;
    for (int it = 0; it < L / 32; ++it) mx = fmaxf(mx, sc[h][it * 32 + lane]);
    for (int o = 16; o; o >>= 1) mx = fmaxf(mx, __shfl_xor(mx, o, 32));
    float sum = 0.f;
    for (int it = 0; it < L / 32; ++it) {
        int l = it * 32 + lane;
        float e = __expf(sc[h][l] - mx);
        sc[h][l] = e;
        sum += e;
    }
    for (int o = 16; o; o >>= 1) sum += __shfl_xor(sum, o, 32);
    const float inv = 1.0f / sum;

    // Phase 3: rank-2 V trick.  out[h,d] = Wv0[hd]*s0 + Wv1[hd]*s1.
    float s0 = 0.f, s1 = 0.f;
    for (int it = 0; it < L / 32; ++it) {
        int l = it * 32 + lane;
        float a = sc[h][l] * inv;
        s0 += a * xs[l * 2];
        s1 += a * xs[l * 2 + 1];
    }
    for (int o = 16; o; o >>= 1) {
        s0 += __shfl_xor(s0, o, 32);
        s1 += __shfl_xor(s1, o, 32);
    }
    const int d = lane;
    last[bp * DMOD + h * 32 + d] = wv[h * 32 + d] * s0 + wv[DMOD + h * 32 + d] * s1;
}

// ---------------------------------------------------------------------------
// fp32 WMMA 16x16 tile GEMM core: one output tile per wave32.
// A: [M,K] row-major, B: [K,N] row-major.
// Lane layout per CDNA5 ISA:
//   A 16x4:  lanes 0-15 rows M=0..15 K={k0,k0+1}; lanes 16-31 same rows K={k0+2,k0+3}
//   B 4x16:  lanes 0-15 cols N=0..15 K={k0,k0+1}; lanes 16-31 K={k0+2,k0+3}
//   C/D:     VGPR j -> row (half*8 + j), col = lane&15
// ---------------------------------------------------------------------------
__device__ inline v8f wmma_tile_f32(const float* __restrict__ A, int lda,
                                    const float* __restrict__ B, int ldb,
                                    int tileM, int tileN, int K)
{
    const int lane = threadIdx.x & 31;
    const int half = lane >> 4;
    const int l16  = lane & 15;
    v8f acc = {0.f, 0.f, 0.f, 0.f, 0.f, 0.f, 0.f, 0.f};
    for (int k0 = 0; k0 < K; k0 += 4) {
        v2f a, b;
        const float* arow = A + (tileM + l16) * lda + k0 + half * 2;
        a.x = arow[0];
        a.y = arow[1];
        b.x = B[(k0 + half * 2 + 0) * ldb + tileN + l16];
        b.y = B[(k0 + half * 2 + 1) * ldb + tileN + l16];
        acc = __builtin_amdgcn_wmma_f32_16x16x4_f32(
            /*neg_a=*/false, a, /*neg_b=*/false, b,
            /*c_mod=*/(short)0, acc, /*reuse_a=*/false, /*reuse_b=*/false);
    }
    return acc;
}

// Kernel 2: H = relu(last @ W1 + b1)   [64,256]x[256,1024]
__global__ __launch_bounds__(256) void gemm1_kernel(
    const float* __restrict__ A, const float* __restrict__ W1,
    const float* __restrict__ b1, float* __restrict__ H)
{
    const int wave = threadIdx.x >> 5;
    const int tile = blockIdx.x * 8 + wave;          // 256 tiles: 4 x 64
    const int tileM = (tile >> 6) * 16;
    const int tileN = (tile & 63) * 16;
    v8f acc = wmma_tile_f32(A, DMOD, W1, DFF, tileM, tileN, DMOD);
    const int lane = threadIdx.x & 31;
    const int half = lane >> 4, l16 = lane & 15;
    const float bias = b1[tileN + l16];
    for (int j = 0; j < 8; ++j) {
        float v = acc[j] + bias;
        H[(tileM + half * 8 + j) * DFF + tileN + l16] = fmaxf(v, 0.f);
    }
}

// Kernel 3: Y = last + H @ W2 + b2     [64,1024]x[1024,256]
__global__ __launch_bounds__(256) void gemm2_kernel(
    const float* __restrict__ H, const float* __restrict__ W2,
    const float* __restrict__ b2, const float* __restrict__ last,
    float* __restrict__ Y)
{
    const int wave = threadIdx.x >> 5;
    const int tile = blockIdx.x * 8 + wave;          // 64 tiles: 4 x 16
    const int tileM = (tile >> 4) * 16;
    const int tileN = (tile & 15) * 16;
    v8f acc = wmma_tile_f32(H, DFF, W2, DMOD, tileM, tileN, DFF);
    const int lane = threadIdx.x & 31;
    const int half = lane >> 4, l16 = lane & 15;
    const float bias = b2[tileN + l16];
    for (int j = 0; j < 8; ++j) {
        int row = tileM + half * 8 + j;
        Y[row * DMOD + tileN + l16] = acc[j] + bias + last[row * DMOD + tileN + l16];
    }
}

// Kernel 4: LayerNorm rows of Y (population variance, eps=1e-5).
__global__ __launch_bounds__(256) void ln_kernel(
    const float* __restrict__ Y, const float* __restrict__ g,
    const float* __restrict__ b, float* __restrict__ Z)
{
    __shared__ float red[8];
    const int row = blockIdx.x;
    const int tid = threadIdx.x;
    const float v = Y[row * DMOD + tid];

    float s = v;
    for (int o = 16; o; o >>= 1) s += __shfl_xor(s, o, 32);
    if ((tid & 31) == 0) red[tid >> 5] = s;
    __syncthreads();
    float tot = 0.f;
    for (int i = 0; i < 8; ++i) tot += red[i];
    const float mu = tot * (1.0f / DMOD);
    __syncthreads();

    float dv = v - mu;
    float s2 = dv * dv;
    for (int o = 16; o; o >>= 1) s2 += __shfl_xor(s2, o, 32);
    if ((tid & 31) == 0) red[tid >> 5] = s2;
    __syncthreads();
    float tot2 = 0.f;
    for (int i = 0; i < 8; ++i) tot2 += red[i];
    const float var = tot2 * (1.0f / DMOD);
    const float rstd = rsqrtf(var + 1e-5f);

    Z[row * DMOD + tid] = dv * rstd * g[tid] + b[tid];
}

// Kernel 5: out = Z @ Wo   [64,256]x[256,512]
__global__ __launch_bounds__(256) void gemm3_kernel(
    const float* __restrict__ Z, const float* __restrict__ Wo,
    float* __restrict__ out)
{
    const int wave = threadIdx.x >> 5;
    const int tile = blockIdx.x * 8 + wave;          // 128 tiles: 4 x 32
    const int tileM = (tile >> 5) * 16;
    const int tileN = (tile & 31) * 16;
    v8f acc = wmma_tile_f32(Z, DMOD, Wo, DM, tileM, tileN, DMOD);
    const int lane = threadIdx.x & 31;
    const int half = lane >> 4, l16 = lane & 15;
    for (int j = 0; j < 8; ++j)
        out[(tileM + half * 8 + j) * DM + tileN + l16] = acc[j];
}

// ---------------------------------------------------------------------------
extern "C" void kernel_launch(void* const* d_in, const int* in_sizes, int n_in,
                              void* d_out, int out_size, void* d_ws, size_t ws_size,
                              hipStream_t stream)
{
    const float* x    = (const float*)d_in[0];
    const float* t    = (const float*)d_in[1];
    const int*   mask = (const int*)  d_in[2];
    const float* emb  = (const float*)d_in[3];
    const float* Wq   = (const float*)d_in[4];
    const float* Wk   = (const float*)d_in[5];
    const float* Wv   = (const float*)d_in[6];
    const float* Wo   = (const float*)d_in[7];
    const float* ln_g = (const float*)d_in[8];
    const float* ln_b = (const float*)d_in[9];
    const float* W1   = (const float*)d_in[10];
    const float* b1   = (const float*)d_in[11];
    const float* W2   = (const float*)d_in[12];
    const float* b2   = (const float*)d_in[13];
    float* out = (float*)d_out;

    float* ws   = (float*)d_ws;
    float* last = ws;                                   // 64*256
    float* h1   = last + BP * DMOD;                     // 64*1024
    float* y    = h1 + BP * DFF;                        // 64*256
    float* z    = y + BP * DMOD;                        // 64*256

    attn_kernel <<<BP, 256, 0, stream>>>(x, t, mask, emb, Wq, Wk, Wv, last);
    gemm1_kernel<<<(4 * (DFF / 16)) / 8, 256, 0, stream>>>(last, W1, b1, h1);
    gemm2_kernel<<<(4 * (DMOD / 16)) / 8, 256, 0, stream>>>(h1, W2, b2, last, y);
    ln_kernel   <<<BP, 256, 0, stream>>>(y, ln_g, ln_b, z);
    gemm3_kernel<<<(4 * (DM / 16)) / 8, 256, 0, stream>>>(z, Wo, out);
}